// MessageBlock_40475771797587
// MI455X (gfx1250) — compile-verified
//
#include <hip/hip_runtime.h>

#define NN   50000
#define NE   800000
#define FDIM 64
#define RDIM 32
#define F3   192
#define NT_E (NE / 16)
#define NT_N (NN / 16)

typedef __attribute__((ext_vector_type(2))) float v2f;
typedef __attribute__((ext_vector_type(8))) float v8f;

typedef __attribute__((address_space(1))) int as1_int;
typedef __attribute__((address_space(3))) int as3_int;

#if defined(__has_builtin)
#if __has_builtin(__builtin_amdgcn_global_load_async_to_lds_b32)
#define HAVE_ASYNC_LDS 1
#endif
#endif

__device__ __forceinline__ v8f wmma_f32(v2f a, v2f b, v8f c) {
    // D = A(16x4 f32) * B(4x16 f32) + C(16x16 f32)
    return __builtin_amdgcn_wmma_f32_16x16x4_f32(
        /*neg_a=*/false, a, /*neg_b=*/false, b,
        /*c_mod=*/(short)0, c, /*reuse_a=*/false, /*reuse_b=*/false);
}

__device__ __forceinline__ v8f splat8(float x) {
    v8f r;
#pragma unroll
    for (int i = 0; i < 8; ++i) r[i] = x;
    return r;
}

__device__ __forceinline__ float silu_fast(float h) {
    // h * sigmoid(h) with fast v_rcp_f32 (activation: no need for IEEE div)
    return h * __builtin_amdgcn_rcpf(1.0f + __expf(-h));
}

// Stage a row-major (rows x cols) matrix into LDS, transposed + row-paired:
//   dstPairs[col*stride + q] = { M[2q][col], M[2q+1][col] }
__device__ __forceinline__ void stage_paired(const float* __restrict__ g,
                                             float* l, int rows, int cols,
                                             int stride) {
    for (int e = threadIdx.x; e < rows * cols; e += 256) {
        const int row = e / cols;
        const int col = e - row * cols;
        float* dst = l + (col * stride + (row >> 1)) * 2 + (row & 1);
#ifdef HAVE_ASYNC_LDS
        __builtin_amdgcn_global_load_async_to_lds_b32(
            (as1_int*)(g + e), (as3_int*)dst, 0, 0);
#else
        *dst = g[e];
#endif
    }
}

__device__ __forceinline__ void stage_fence() {
#ifdef HAVE_ASYNC_LDS
    asm volatile("s_wait_asynccnt 0" ::: "memory");
#endif
    __syncthreads();
}

// ---------------------------------------------------------------------------
// phi = silu(s @ W1 + b1) @ W2 + b2        (per wave: 16-node row tiles,
// grid-stride; W1/W2 LDS-resident in paired layout)
// ---------------------------------------------------------------------------
__global__ void __launch_bounds__(256) phi_kernel(
    const float* __restrict__ s,
    const float* __restrict__ W1, const float* __restrict__ b1,
    const float* __restrict__ W2, const float* __restrict__ b2,
    float* __restrict__ phi) {

    __shared__ v2f   w1p[FDIM * 35];    // 17920 B : W1 (64x64) paired
    __shared__ v2f   w2p[F3 * 35];      // 53760 B : W2 (64x192) paired
    __shared__ float hsh[8][16 * 68];   // 34816 B : per-wave SiLU tile

    stage_paired(W1, (float*)w1p, FDIM, FDIM, 35);
    stage_paired(W2, (float*)w2p, FDIM, F3, 35);
    stage_fence();

    const int lane = threadIdx.x & 31;
    const int wv   = threadIdx.x >> 5;
    const int lo   = lane & 15;         // 0..15
    const int hi   = lane >> 4;         // 0 or 1

    float* hbuf = hsh[wv];

    // Tile-invariant biases (per-lane column values)
    float b1v[4], b2v[12];
#pragma unroll
    for (int n = 0; n < 4; ++n)  b1v[n] = b1[n * 16 + lo];
#pragma unroll
    for (int n = 0; n < 12; ++n) b2v[n] = b2[n * 16 + lo];

    const int wstride = gridDim.x * 8;
    for (int tile = blockIdx.x * 8 + wv; tile < NT_N; tile += wstride) {
        const int r0 = tile * 16;

        // A-fragments of s tile: 16 k-frags of 16x4 (row=lo, contiguous f2)
        v2f a1[16];
        const float* srow = s + (size_t)(r0 + lo) * FDIM + 2 * hi;
#pragma unroll
        for (int k = 0; k < 16; ++k)
            a1[k] = *(const v2f*)(srow + 4 * k);

        // GEMM1 (K=64, N=64) + SiLU, C-layout -> LDS
#pragma unroll
        for (int n = 0; n < 4; ++n) {
            v8f acc = splat8(b1v[n]);
            const int bcol = (n * 16 + lo) * 35;
#pragma unroll
            for (int k = 0; k < 16; ++k)
                acc = wmma_f32(a1[k], w1p[bcol + 2 * k + hi], acc);
#pragma unroll
            for (int g = 0; g < 8; ++g)
                hbuf[(g + 8 * hi) * 68 + n * 16 + lo] = silu_fast(acc[g]);
        }

        // Same-wave LDS RAW: DS ops in-order per wave; fence codegen order.
        asm volatile("s_wait_dscnt 0" ::: "memory");

        // Re-read SiLU tile in A-fragment layout (conflict-free, stride 68)
        v2f a2[16];
#pragma unroll
        for (int k = 0; k < 16; ++k) {
            const int base = lo * 68 + 4 * k + 2 * hi;
            v2f t = { hbuf[base], hbuf[base + 1] };
            a2[k] = t;
        }

        // GEMM2 (K=64, N=192) -> phi
#pragma unroll
        for (int n = 0; n < 12; ++n) {
            v8f acc = splat8(b2v[n]);
            const int bcol = (n * 16 + lo) * 35;
#pragma unroll
            for (int k = 0; k < 16; ++k)
                acc = wmma_f32(a2[k], w2p[bcol + 2 * k + hi], acc);
#pragma unroll
            for (int g = 0; g < 8; ++g)
                phi[(size_t)(r0 + g + 8 * hi) * F3 + n * 16 + lo] = acc[g];
        }
    }
}

// ---------------------------------------------------------------------------
// Edge kernel: W = (radial@Wr + br)*f_cut (WMMA on LDS-resident Wr),
// fused gather (phi[j], v[j]) / scatter (atomic f32) epilogue.
// ---------------------------------------------------------------------------
__global__ void __launch_bounds__(256) edge_kernel(
    const float* __restrict__ radial, const float* __restrict__ fcut,
    const float* __restrict__ uvec,   const int*   __restrict__ eidx,
    const float* __restrict__ Wr,     const float* __restrict__ br,
    const float* __restrict__ phi,    const float* __restrict__ v,
    float* __restrict__ out_s,        float* __restrict__ out_v) {

    __shared__ v2f wr2[F3 * 17];        // 26112 B : Wr (32x192) paired

    stage_paired(Wr, (float*)wr2, RDIM, F3, 17);
    stage_fence();

    const int lane = threadIdx.x & 31;
    const int wv   = threadIdx.x >> 5;
    const int lo   = lane & 15;
    const int hi   = lane >> 4;

    // Tile-invariant bias values for this lane's columns
    float br0[4], brv[4], brs[4];
#pragma unroll
    for (int c = 0; c < 4; ++c) {
        br0[c] = br[c * 16 + lo];
        brv[c] = br[64 + c * 16 + lo];
        brs[c] = br[128 + c * 16 + lo];
    }

    const int wstride = gridDim.x * 8;

    for (int tile = blockIdx.x * 8 + wv; tile < NT_E; tile += wstride) {
        const int e0 = tile * 16;

        // Prefetch next tile's radial rows (global_prefetch_b8)
        if (tile + wstride < NT_E)
            __builtin_prefetch(
                radial + (size_t)((tile + wstride) * 16 + lo) * RDIM, 0, 1);

        // A-fragments of radial tile (16x32): 8 k-frags, float2 per lane
        v2f a[8];
        const float* rrow = radial + (size_t)(e0 + lo) * RDIM + 2 * hi;
#pragma unroll
        for (int k = 0; k < 8; ++k)
            a[k] = *(const v2f*)(rrow + 4 * k);

        // Per-lane row metadata: this lane's C rows are e0 + 8*hi + g
        const int base = e0 + 8 * hi;
        int   ii[8], ji[8];
        float fc[8];
#pragma unroll
        for (int g = 0; g < 8; ++g) {
            ii[g] = eidx[base + g];            // idx_i
            ji[g] = eidx[NE + base + g];       // idx_j
            fc[g] = fcut[base + g];
        }
        // unit vectors for this lane's 8 rows (24 contiguous floats)
        float uu[24];
#pragma unroll
        for (int t = 0; t < 24; ++t)
            uu[t] = uvec[(size_t)base * 3 + t];

        // ---- scalar message: cols 0..63 -> atomic add into out_s ----
#pragma unroll
        for (int c = 0; c < 4; ++c) {
            const int col = c * 16 + lo;
            v8f acc = splat8(br0[c]);
#pragma unroll
            for (int k = 0; k < 8; ++k)
                acc = wmma_f32(a[k], wr2[col * 17 + 2 * k + hi], acc);
#pragma unroll
            for (int g = 0; g < 8; ++g) {
                float xs = acc[g] * fc[g] * phi[(size_t)ji[g] * F3 + col];
                __hip_atomic_fetch_add(out_s + (size_t)ii[g] * FDIM + col, xs,
                                       __ATOMIC_RELAXED,
                                       __HIP_MEMORY_SCOPE_AGENT);
            }
        }

        // ---- vector message: x_vv (cols 64..127), x_vs (cols 128..191) ----
#pragma unroll
        for (int c = 0; c < 4; ++c) {
            const int col  = c * 16 + lo;       // feature col 0..63
            const int colv = 64  + col;         // x_vv col
            const int cols = 128 + col;         // x_vs col

            v8f avv = splat8(brv[c]);
            v8f avs = splat8(brs[c]);
#pragma unroll
            for (int k = 0; k < 8; ++k)
                avv = wmma_f32(a[k], wr2[colv * 17 + 2 * k + hi], avv);
#pragma unroll
            for (int k = 0; k < 8; ++k)
                avs = wmma_f32(a[k], wr2[cols * 17 + 2 * k + hi], avs);

            float xvv[8], xvs[8];
#pragma unroll
            for (int g = 0; g < 8; ++g) {
                const size_t jb = (size_t)ji[g] * F3;
                xvv[g] = avv[g] * fc[g] * phi[jb + colv];
                xvs[g] = avs[g] * fc[g] * phi[jb + cols];
            }

#pragma unroll
            for (int d = 0; d < 3; ++d) {
#pragma unroll
                for (int g = 0; g < 8; ++g) {
                    float vj = v[(size_t)ji[g] * F3 + d * FDIM + col];
                    float xv = vj * xvv[g] + xvs[g] * uu[g * 3 + d];
                    __hip_atomic_fetch_add(
                        out_v + (size_t)ii[g] * F3 + d * FDIM + col, xv,
                        __ATOMIC_RELAXED, __HIP_MEMORY_SCOPE_AGENT);
                }
            }
        }
    }
}

// ---------------------------------------------------------------------------
extern "C" void kernel_launch(void* const* d_in, const int* in_sizes, int n_in,
                              void* d_out, int out_size, void* d_ws, size_t ws_size,
                              hipStream_t stream) {
    (void)in_sizes; (void)n_in; (void)out_size; (void)ws_size;

    const float* s      = (const float*)d_in[0];
    const float* v      = (const float*)d_in[1];
    const float* radial = (const float*)d_in[2];
    const float* fcut   = (const float*)d_in[3];
    const float* uvec   = (const float*)d_in[4];
    const int*   eidx   = (const int*)  d_in[5];
    const float* W1     = (const float*)d_in[6];
    const float* b1     = (const float*)d_in[7];
    const float* W2     = (const float*)d_in[8];
    const float* b2     = (const float*)d_in[9];
    const float* Wr     = (const float*)d_in[10];
    const float* br     = (const float*)d_in[11];

    float* out_s = (float*)d_out;
    float* out_v = out_s + (size_t)NN * FDIM;
    float* phi   = (float*)d_ws;                  // 50000*192*4 = 38.4 MB

    // Seed outputs with s and v (outputs are s+ds, v+dv)
    (void)hipMemcpyAsync(out_s, s, (size_t)NN * FDIM * sizeof(float),
                         hipMemcpyDeviceToDevice, stream);
    (void)hipMemcpyAsync(out_v, v, (size_t)NN * F3 * sizeof(float),
                         hipMemcpyDeviceToDevice, stream);

    // 3125 node tiles, grid-stride over 96 blocks x 8 waves
    phi_kernel<<<96, 256, 0, stream>>>(s, W1, b1, W2, b2, phi);

    // 50000 edge tiles, grid-stride over 512 blocks x 8 waves
    edge_kernel<<<512, 256, 0, stream>>>(
        radial, fcut, uvec, eidx, Wr, br, phi, v, out_s, out_v);
}